// HeteroLightGCN_51719996178617
// MI455X (gfx1250) — compile-verified
//
#include <hip/hip_runtime.h>
#include <hip/hip_bf16.h>

// ---------------------------------------------------------------------------
// HeteroLightGCN on MI455X (gfx1250, wave32)
//   u0 = user_feat @ W_user ; b0 = biz_feat @ W_biz        (WMMA f32 16x16x4)
//   3 layers of SpMM scatter-add (native global_atomic_add_f32, edge-per-wave)
//   mean over (layers+1) then row-wise L2 normalize (wave32 shfl reduction)
// ---------------------------------------------------------------------------

#define IN_DIM   384
#define EMB      128
#define NLAYERS  3
#define LDA      388   // 16-row LDS tile stride, padded: 388 % 64 banks == 4 -> conflict-free

typedef float v2f __attribute__((ext_vector_type(2)));
typedef float v8f __attribute__((ext_vector_type(8)));

// Native CDNA5 fp32 global atomic add, non-returning (TH[0]=0 -> STOREcnt),
// device scope so cross-workgroup scatter-adds are coherent at L2.
// Inline asm guarantees we never fall back to a CAS expansion loop.
__device__ __forceinline__ void atomic_add_f32_dev(float* p, float v) {
#if defined(__gfx1250__)
    asm volatile("global_atomic_add_f32 %0, %1, off scope:SCOPE_DEV"
                 :
                 : "v"(p), "v"(v)
                 : "memory");
#else
    (void)__hip_atomic_fetch_add(p, v, __ATOMIC_RELAXED, __HIP_MEMORY_SCOPE_AGENT);
#endif
}

// ---------------------------------------------------------------------------
// Dense GEMM: C[n_rows x 128] = A[n_rows x 384] @ W[384 x 128], fp32 WMMA.
// Block = 256 threads = 8 waves. Block computes 16 rows x 128 cols.
// Wave w owns N-tile [w*16, w*16+16). K-loop: 96 steps of 16x16x4.
// n_rows must be a multiple of 16 (100000 and 20000 both are).
// Writes result to both Ccur and Csum (sum is initialized with layer-0 term).
// ---------------------------------------------------------------------------
__global__ __launch_bounds__(256)
void gemm_wmma_f32(const float* __restrict__ A, const float* __restrict__ W,
                   float* __restrict__ Ccur, float* __restrict__ Csum) {
    __shared__ float ldsA[16 * LDA];

    const int m0  = blockIdx.x * 16;
    const int tid = threadIdx.x;

    // Cooperative, coalesced load of the 16x384 A tile into LDS.
    for (int i = tid; i < 16 * IN_DIM; i += 256) {
        int r = i / IN_DIM;
        int c = i - r * IN_DIM;
        ldsA[r * LDA + c] = A[(size_t)(m0 + r) * IN_DIM + c];
    }
    __syncthreads();

    const int wave = tid >> 5;
    const int lane = tid & 31;
    const int n0   = wave * 16;
    const int half = lane >> 4;   // 0: lanes 0-15, 1: lanes 16-31
    const int l16  = lane & 15;

    v8f c = {};
    for (int kk = 0; kk < IN_DIM; kk += 4) {
        const int ka = kk + 2 * half;   // lanes 0-15 see K {kk,kk+1}; lanes 16-31 {kk+2,kk+3}
        v2f a, b;
        // A fragment (16x4): lane L<16 row=L holds K=ka..ka+1 in VGPR0/1
        a.x = ldsA[l16 * LDA + ka];
        a.y = ldsA[l16 * LDA + ka + 1];
        // B fragment (4x16): VGPR0 lanes0-15 K=kk row, lanes16-31 K=kk+2; VGPR1 = +1
        b.x = W[(size_t)(ka)     * EMB + n0 + l16];
        b.y = W[(size_t)(ka + 1) * EMB + n0 + l16];
        c = __builtin_amdgcn_wmma_f32_16x16x4_f32(
                /*neg_a=*/false, a, /*neg_b=*/false, b,
                /*c_mod=*/(short)0, c, /*reuse_a=*/false, /*reuse_b=*/false);
    }

    // C/D layout: VGPR j, lanes 0-15 -> (M=j, N=lane); lanes 16-31 -> (M=8+j, N=lane-16)
    const int mbase = m0 + half * 8;
    const int ncol  = n0 + l16;
#pragma unroll
    for (int j = 0; j < 8; ++j) {
        size_t off = (size_t)(mbase + j) * EMB + ncol;
        float  v   = c[j];
        Ccur[off] = v;
        Csum[off] = v;
    }
}

// ---------------------------------------------------------------------------
// SpMM scatter-add: out[dst[e], :] += val[e] * x[src[e], :]
// One wave32 per edge; each lane handles a float4 (32 lanes * 4 = 128 dims).
// src/dst/val loads are wave-uniform (scalarized); gather is a coalesced 512B
// read that stays L2-resident (x arrays are 51MB/10MB << 192MB L2).
// ---------------------------------------------------------------------------
__global__ __launch_bounds__(256)
void spmm_edge(const float* __restrict__ x, const int* __restrict__ src,
               const int* __restrict__ dst, const float* __restrict__ val,
               float* __restrict__ out, int n_edges) {
    const int e = (int)((blockIdx.x * 256u + threadIdx.x) >> 5);
    if (e >= n_edges) return;
    const int lane = threadIdx.x & 31;

    const int   s = src[e];
    const int   d = dst[e];
    const float w = val[e];

    const float4 v = ((const float4*)(x + (size_t)s * EMB))[lane];
    float* o = out + (size_t)d * EMB + lane * 4;

    atomic_add_f32_dev(o + 0, v.x * w);
    atomic_add_f32_dev(o + 1, v.y * w);
    atomic_add_f32_dev(o + 2, v.z * w);
    atomic_add_f32_dev(o + 3, v.w * w);
}

// acc += x, float4-vectorized
__global__ __launch_bounds__(256)
void add_inplace(float* __restrict__ acc, const float* __restrict__ x, int n4) {
    int i = blockIdx.x * 256 + threadIdx.x;
    if (i >= n4) return;
    float4 a = ((float4*)acc)[i];
    const float4 b = ((const float4*)x)[i];
    a.x += b.x; a.y += b.y; a.z += b.z; a.w += b.w;
    ((float4*)acc)[i] = a;
}

// out[row,:] = l2norm(sum[row,:] * scale). One wave32 per 128-wide row.
__global__ __launch_bounds__(256)
void finalize_l2(const float* __restrict__ sum, float* __restrict__ out,
                 int n_rows, float scale) {
    const int row = (int)((blockIdx.x * 256u + threadIdx.x) >> 5);
    if (row >= n_rows) return;
    const int lane = threadIdx.x & 31;

    float4 v = ((const float4*)(sum + (size_t)row * EMB))[lane];
    v.x *= scale; v.y *= scale; v.z *= scale; v.w *= scale;

    float ss = v.x * v.x + v.y * v.y + v.z * v.z + v.w * v.w;
#pragma unroll
    for (int m = 16; m > 0; m >>= 1)
        ss += __shfl_xor(ss, m, 32);          // wave32 butterfly reduction

    const float nrm = sqrtf(ss);
    const float inv = 1.0f / fmaxf(nrm, 1e-12f);
    v.x *= inv; v.y *= inv; v.z *= inv; v.w *= inv;
    ((float4*)(out + (size_t)row * EMB))[lane] = v;
}

// ---------------------------------------------------------------------------
extern "C" void kernel_launch(void* const* d_in, const int* in_sizes, int n_in,
                              void* d_out, int out_size, void* d_ws, size_t ws_size,
                              hipStream_t stream) {
    const float* user_feat = (const float*)d_in[0];
    const float* biz_feat  = (const float*)d_in[1];
    const float* W_user    = (const float*)d_in[2];
    const float* W_biz     = (const float*)d_in[3];
    const int*   ub_u      = (const int*)  d_in[4];
    const int*   ub_b      = (const int*)  d_in[5];
    const float* val_ub    = (const float*)d_in[6];
    const float* val_bu    = (const float*)d_in[7];
    const int*   uu_src    = (const int*)  d_in[8];
    const int*   uu_dst    = (const int*)  d_in[9];
    const float* uu_val    = (const float*)d_in[10];
    const int*   bb_src    = (const int*)  d_in[11];
    const int*   bb_dst    = (const int*)  d_in[12];
    const float* bb_val    = (const float*)d_in[13];

    const int n_u  = in_sizes[0] / IN_DIM;   // 100000
    const int n_b  = in_sizes[1] / IN_DIM;   // 20000
    const int e_ub = in_sizes[4];            // 3.2M
    const int e_uu = in_sizes[8];            // 1.6M
    const int e_bb = in_sizes[11];           // 320K

    const size_t NU = (size_t)n_u * EMB;     // floats per user matrix
    const size_t NB = (size_t)n_b * EMB;

    // Workspace layout (fp32): u_cur | u_nxt | u_sum | b_cur | b_nxt | b_sum
    float* ws    = (float*)d_ws;
    float* u_cur = ws;
    float* u_nxt = ws + NU;
    float* u_sum = ws + 2 * NU;
    float* b_cur = ws + 3 * NU;
    float* b_nxt = ws + 3 * NU + NB;
    float* b_sum = ws + 3 * NU + 2 * NB;

    float* out_u = (float*)d_out;
    float* out_b = out_u + NU;

    // Layer 0: dense projections (also seeds the running sums).
    gemm_wmma_f32<<<n_u / 16, 256, 0, stream>>>(user_feat, W_user, u_cur, u_sum);
    gemm_wmma_f32<<<n_b / 16, 256, 0, stream>>>(biz_feat,  W_biz,  b_cur, b_sum);

    const int EPB = 8;  // edges per 256-thread block (1 edge per wave32)
    for (int layer = 0; layer < NLAYERS; ++layer) {
        hipMemsetAsync(u_nxt, 0, NU * sizeof(float), stream);
        hipMemsetAsync(b_nxt, 0, NB * sizeof(float), stream);

        // u_new = adj_bu @ b_cur + adj_uu @ u_cur
        spmm_edge<<<(e_ub + EPB - 1) / EPB, 256, 0, stream>>>(b_cur, ub_b, ub_u, val_bu, u_nxt, e_ub);
        spmm_edge<<<(e_uu + EPB - 1) / EPB, 256, 0, stream>>>(u_cur, uu_src, uu_dst, uu_val, u_nxt, e_uu);
        // b_new = adj_ub @ u_cur + adj_bb @ b_cur
        spmm_edge<<<(e_ub + EPB - 1) / EPB, 256, 0, stream>>>(u_cur, ub_u, ub_b, val_ub, b_nxt, e_ub);
        spmm_edge<<<(e_bb + EPB - 1) / EPB, 256, 0, stream>>>(b_cur, bb_src, bb_dst, bb_val, b_nxt, e_bb);

        add_inplace<<<(int)((NU / 4 + 255) / 256), 256, 0, stream>>>(u_sum, u_nxt, (int)(NU / 4));
        add_inplace<<<(int)((NB / 4 + 255) / 256), 256, 0, stream>>>(b_sum, b_nxt, (int)(NB / 4));

        // Host-side double-buffer swap (deterministic; safe under graph capture).
        float* t;
        t = u_cur; u_cur = u_nxt; u_nxt = t;
        t = b_cur; b_cur = b_nxt; b_nxt = t;
    }

    const float scale = 1.0f / (NLAYERS + 1);   // 0.25
    finalize_l2<<<(n_u * 32 + 255) / 256, 256, 0, stream>>>(u_sum, out_u, n_u, scale);
    finalize_l2<<<(n_b * 32 + 255) / 256, 256, 0, stream>>>(b_sum, out_b, n_b, scale);
}